// SphericalHarmonicEdgeAttrs_59588376265204
// MI455X (gfx1250) — compile-verified
//
#include <hip/hip_runtime.h>
#include <hip/hip_bf16.h>

#define BLOCK 256

// gfx1250 async global->LDS path (ASYNCcnt). Guarded: if the builtins are
// absent (or host pass), fall back to plain coalesced loads.
#if defined(__AMDGCN__) && __has_builtin(__builtin_amdgcn_global_load_async_to_lds_b32) && __has_builtin(__builtin_amdgcn_s_wait_asynccnt)
#define USE_ASYNC 1
#else
#define USE_ASYNC 0
#endif

typedef float v4f __attribute__((ext_vector_type(4)));

__global__ __launch_bounds__(BLOCK) void sh_edge_kernel(
    const float* __restrict__ pos,
    const int* __restrict__ ei,   // [2, nE]: row0 = sender, row1 = receiver
    float* __restrict__ out,      // [nE, 16]
    int nE, int ntiles)
{
  const int tid = threadIdx.x;
  const int* __restrict__ esrc = ei;
  const int* __restrict__ edst = ei + nE;

#if USE_ASYNC
  __shared__ int s_s[2][BLOCK];
  __shared__ int s_r[2][BLOCK];
#endif

  int tile = blockIdx.x;
  const int stride = gridDim.x;

#if USE_ASYNC
  {
    int e0 = tile * BLOCK + tid;
    int ec = e0 < nE ? e0 : (nE - 1);
    __builtin_amdgcn_global_load_async_to_lds_b32((int*)(esrc + ec), &s_s[0][tid], 0, 0);
    __builtin_amdgcn_global_load_async_to_lds_b32((int*)(edst + ec), &s_r[0][tid], 0, 0);
  }
  int buf = 0;
#endif

  for (; tile < ntiles; tile += stride) {
    const int e = tile * BLOCK + tid;
    int is, ir;

#if USE_ASYNC
    const int nt = tile + stride;
    if (nt < ntiles) {
      // prefetch next tile's indices into the other LDS buffer
      int e1 = nt * BLOCK + tid;
      int ec = e1 < nE ? e1 : (nE - 1);
      __builtin_amdgcn_global_load_async_to_lds_b32((int*)(esrc + ec), &s_s[buf ^ 1][tid], 0, 0);
      __builtin_amdgcn_global_load_async_to_lds_b32((int*)(edst + ec), &s_r[buf ^ 1][tid], 0, 0);
      // allow the 2 just-issued loads to remain outstanding; current buf done
      __builtin_amdgcn_s_wait_asynccnt(2);
    } else {
      __builtin_amdgcn_s_wait_asynccnt(0);
    }
    // compiler barrier so the LDS reads below stay after the asynccnt wait;
    // plain shared-array access keeps addrspace(3) => ds_load_b32 (not flat)
    asm volatile("" ::: "memory");
    is = s_s[buf][tid];
    ir = s_r[buf][tid];
    buf ^= 1;
#else
    const int ec = e < nE ? e : (nE - 1);
    is = esrc[ec];
    ir = edst[ec];
#endif

    // gathers: 1.2 MB table -> L2-resident
    const float* ps = pos + 3 * (size_t)(unsigned)is;
    const float* pr = pos + 3 * (size_t)(unsigned)ir;
    float x = pr[0] - ps[0];
    float y = pr[1] - ps[1];
    float z = pr[2] - ps[2];

    // F.normalize semantics: v / max(||v||, 1e-12)
    float n2  = fmaf(x, x, fmaf(y, y, z * z));
    float nrm = sqrtf(n2);
    float inv = 1.0f / fmaxf(nrm, 1e-12f);
    x *= inv; y *= inv; z *= inv;

    const float y2   = y * y;
    const float x2   = x * x;
    const float z2   = z * z;
    const float x2z2 = x2 + z2;
    const float a    = x * z;                   // xz
    const float b    = z2 - x2;                 // z^2 - x^2
    const float q    = fmaf(4.0f, y2, -x2z2);   // 4y^2 - (x^2+z^2)

    // 'component' normalization folded into constants: l0:1, l1:sqrt3, l2:sqrt5, l3:sqrt7
    v4f o0, o1, o2, o3;
    o0.x = 1.0f;
    o0.y = 1.7320508075688772f * x;                       // sqrt3 * x
    o0.z = 1.7320508075688772f * y;
    o0.w = 1.7320508075688772f * z;

    o1.x = 3.872983346207417f * a;                        // sqrt15 * xz
    o1.y = 3.872983346207417f * (x * y);                  // sqrt15 * xy
    o1.z = 2.23606797749979f * fmaf(-0.5f, x2z2, y2);     // sqrt5 * (y^2 - 0.5(x^2+z^2))
    o1.w = 3.872983346207417f * (y * z);                  // sqrt15 * yz

    o2.x = 1.9364916731037085f * b;                       // (sqrt15/2) * (z^2-x^2)
    o2.y = 2.4748737341529163f * fmaf(2.0f * a, z, b * x);// 1.75*sqrt2*(2xz*z + b*x)
    o2.z = 12.12435565298214f  * (a * y);                 // 7*sqrt3 * xyz
    o2.w = 4.2866070498705616f * (q * x);                 // 1.75*sqrt6 * (4y^2-x2z2)*x

    o3.x = 3.5f * (y * fmaf(2.0f, y2, -3.0f * x2z2));     // 3.5*y*(2y^2-3x2z2)
    o3.y = 4.2866070498705616f * (q * z);                 // 1.75*sqrt6 * z*(4y^2-x2z2)
    o3.z = 6.062177826491071f  * (b * y);                 // (7*sqrt3/2) * b*y
    o3.w = 2.4748737341529163f * fmaf(-2.0f * a, x, b * z);// 1.75*sqrt2*(b*z - 2xz*x)

    if (e < nE) {
      // 410 MB write-once stream: non-temporal 128-bit stores
      v4f* op = (v4f*)(out + (size_t)e * 16);
      __builtin_nontemporal_store(o0, op + 0);
      __builtin_nontemporal_store(o1, op + 1);
      __builtin_nontemporal_store(o2, op + 2);
      __builtin_nontemporal_store(o3, op + 3);
    }
  }
}

extern "C" void kernel_launch(void* const* d_in, const int* in_sizes, int n_in,
                              void* d_out, int out_size, void* d_ws, size_t ws_size,
                              hipStream_t stream) {
  const float* pos = (const float*)d_in[0];          // [N_NODES, 3] f32
  const int*   ei  = (const int*)d_in[1];            // [2, N_EDGES] i32
  float*       out = (float*)d_out;                  // [N_EDGES, 16] f32

  const int nE     = in_sizes[1] / 2;
  const int ntiles = (nE + BLOCK - 1) / BLOCK;
  int grid = ntiles < 4096 ? ntiles : 4096;          // persistent blocks -> async pipeline depth
  if (grid < 1) grid = 1;

  sh_edge_kernel<<<grid, BLOCK, 0, stream>>>(pos, ei, out, nE, ntiles);
}